// EC_87205015978654
// MI455X (gfx1250) — compile-verified
//
#include <hip/hip_runtime.h>
#include <math.h>

typedef __attribute__((ext_vector_type(2))) float v2f;
typedef __attribute__((ext_vector_type(8))) float v8f;

#define N_NODES 100000
#define IN_CH   256
#define OUT_CH  64
#define N_EDGES 400000
#define ALPHA_F 0.25f

// Workspace layout (in floats):
//  [0, 4096)               B operand, packed per WMMA lane layout: [64 ksteps][32 lanes][2]
//  [4096, 4096+200000)     s[node][2] = { x[n]·A0 , x[n]·A1 }
//  [204096, 204098)        accumulators { sum(loss*m), sum(m) }
#define WS_B   0
#define WS_S   4096
#define WS_ACC (4096 + 2 * N_NODES)

// ---------------------------------------------------------------------------
// Kernel 1: fold W_f into W_d:  A = W_d^T [u v],  u = wa+wc, v = wb-wc.
// Emit A directly in the V_WMMA_F32_16X16X4_F32 B-matrix VGPR layout:
//   for k-step kk (k = 4*kk): lanes 0-15 hold {B[k][n], B[k+1][n]},
//                             lanes 16-31 hold {B[k+2][n], B[k+3][n]}.
// Columns 2..15 are zero padding. One block, 256 threads (thread = k).
// ---------------------------------------------------------------------------
__global__ void __launch_bounds__(256)
prep_kernel(const float* __restrict__ Wd, const float* __restrict__ Wf,
            float* __restrict__ ws) {
    __shared__ float ush[OUT_CH];
    __shared__ float vsh[OUT_CH];
    const int t = threadIdx.x;
    if (t < OUT_CH) {
        ush[t] = Wf[t] + Wf[2 * OUT_CH + t];
        vsh[t] = Wf[OUT_CH + t] - Wf[2 * OUT_CH + t];
    }
    __syncthreads();

    float su = 0.f, sv = 0.f;
    #pragma unroll 8
    for (int o = 0; o < OUT_CH; ++o) {
        const float w = Wd[o * IN_CH + t];   // W_d is [out, in] row-major
        su = fmaf(w, ush[o], su);
        sv = fmaf(w, vsh[o], sv);
    }

    const int kk       = t >> 2;
    const int koff     = t & 3;
    const int laneBase = (koff < 2) ? 0 : 16;
    const int slot     = koff & 1;
    float* bp = ws + WS_B;
    #pragma unroll
    for (int col = 0; col < 16; ++col) {
        const float val = (col == 0) ? su : (col == 1) ? sv : 0.f;
        bp[(kk * 32 + laneBase + col) * 2 + slot] = val;
    }
    if (t == 0) { ws[WS_ACC] = 0.f; ws[WS_ACC + 1] = 0.f; }
}

// ---------------------------------------------------------------------------
// Kernel 2: s[n][0..1] = x[n] · A   via fp32 WMMA 16x16x4.
// One wave per 16 rows of x; 64 WMMA steps over K=256. Memory bound:
// streams 102.4 MB of x (~4.4 us at 23.3 TB/s HBM).
// ---------------------------------------------------------------------------
__global__ void __launch_bounds__(256)
matvec_wmma_kernel(const float* __restrict__ x, float* __restrict__ ws) {
    const int wave   = (int)((blockIdx.x * blockDim.x + threadIdx.x) >> 5);
    const int nWaves = N_NODES / 16;            // 6250, exact
    if (wave >= nWaves) return;                 // wave-uniform: EXEC stays all-1s
    const int lane = threadIdx.x & 31;
    const int row0 = wave * 16;

    // A-matrix 16x4 fp32 layout: lanes 0-15 -> {K0,K1}, lanes 16-31 -> {K2,K3}.
    const int    rowA = row0 + (lane & 15);
    const float* ax   = x + (size_t)rowA * IN_CH + ((lane & 16) ? 2 : 0);
    const v2f*   bp   = (const v2f*)(ws + WS_B);

    v8f c = {};
    #pragma unroll 8
    for (int kk = 0; kk < 64; ++kk) {
        v2f a = *(const v2f*)(ax + 4 * kk);     // global_load_b64, 16B-tile of row
        v2f b = bp[kk * 32 + lane];             // coalesced b64, hot in cache
        c = __builtin_amdgcn_wmma_f32_16x16x4_f32(
                /*neg_a=*/false, a, /*neg_b=*/false, b,
                /*c_mod=*/(short)0, c, /*reuse_a=*/false, /*reuse_b=*/false);
    }

    // C/D layout: VGPR r, lanes 0-15 -> (M=r, N=lane); lanes 16-31 -> (M=8+r).
    // Only output columns 0 (s_u) and 1 (s_v) are meaningful.
    const int col = lane & 15;
    if (col < 2) {
        float* s = ws + WS_S;
        const int rbase = row0 + ((lane & 16) ? 8 : 0);
        #pragma unroll
        for (int r = 0; r < 8; ++r) s[(size_t)(rbase + r) * 2 + col] = c[r];
    }
}

// ---------------------------------------------------------------------------
// Kernel 3: per-edge score + focal loss, masked block reduction.
// ---------------------------------------------------------------------------
__global__ void __launch_bounds__(256)
edge_kernel(const int* __restrict__ eidx, const int* __restrict__ labels,
            const int* __restrict__ mask, const float* __restrict__ ws,
            float* __restrict__ out, float* __restrict__ acc) {
    const int e   = (int)(blockIdx.x * blockDim.x + threadIdx.x);
    const int tid = threadIdx.x;
    const float* s = ws + WS_S;

    float lm = 0.f, cm = 0.f;
    if (e < N_EDGES) {
        const int src = eidx[e];
        const int dst = eidx[N_EDGES + e];
        const float pre   = s[(size_t)src * 2 + 0] + s[(size_t)dst * 2 + 1];
        const float score = 1.f / (1.f + __expf(-pre));
        out[e] = score;
        // reference quirk: sigmoid applied AGAIN to the sigmoided score
        const float p     = 1.f / (1.f + __expf(-score));
        const bool  pos   = (labels[e] == 1);
        const float pt    = pos ? p : (1.f - p);
        const float logpt = __logf(pt);
        const float at    = pos ? (1.f - ALPHA_F) : ALPHA_F;
        const float omp   = 1.f - pt;
        const float loss  = -(omp * omp) * (logpt * at);
        if (mask[e] != 0) { lm = loss; cm = 1.f; }
    }

    __shared__ float rl[256];
    __shared__ float rc[256];
    rl[tid] = lm; rc[tid] = cm;
    __syncthreads();
    for (int stride = 128; stride > 0; stride >>= 1) {
        if (tid < stride) { rl[tid] += rl[tid + stride]; rc[tid] += rc[tid + stride]; }
        __syncthreads();
    }
    if (tid == 0) {
        atomicAdd(&acc[0], rl[0]);
        atomicAdd(&acc[1], rc[0]);
    }
}

__global__ void finalize_kernel(const float* __restrict__ acc, float* __restrict__ out) {
    out[N_EDGES] = acc[0] / acc[1];
}

// ---------------------------------------------------------------------------
extern "C" void kernel_launch(void* const* d_in, const int* in_sizes, int n_in,
                              void* d_out, int out_size, void* d_ws, size_t ws_size,
                              hipStream_t stream) {
    const float* x      = (const float*)d_in[0];
    const float* Wd     = (const float*)d_in[1];
    const float* Wf     = (const float*)d_in[2];
    const int*   eidx   = (const int*)d_in[3];
    const int*   labels = (const int*)d_in[4];
    const int*   mask   = (const int*)d_in[5];
    float*       out    = (float*)d_out;
    float*       ws     = (float*)d_ws;
    float*       acc    = ws + WS_ACC;

    prep_kernel<<<1, 256, 0, stream>>>(Wd, Wf, ws);

    const int nWaves = N_NODES / 16;                  // 6250
    const int nBlk   = (nWaves + 7) / 8;              // 8 waves (256 thr) / block
    matvec_wmma_kernel<<<nBlk, 256, 0, stream>>>(x, ws);

    edge_kernel<<<(N_EDGES + 255) / 256, 256, 0, stream>>>(eidx, labels, mask, ws, out, acc);

    finalize_kernel<<<1, 1, 0, stream>>>(acc, out);
}